// ChamferLoss_11544872092235
// MI455X (gfx1250) — compile-verified
//
#include <hip/hip_runtime.h>
#include <math.h>
#include <stdint.h>

typedef float v2f __attribute__((ext_vector_type(2)));
typedef float v8f __attribute__((ext_vector_type(8)));

#define B_  8
#define N_  2048
#define M_  2048
#define NT  (N_ / 16)       // 128 row tiles per batch
#define MT  (M_ / 16)       // 128 col tiles per batch
#define WAVES_PER_BLOCK 8
#define BLOCKS_PER_BATCH (NT / WAVES_PER_BLOCK)   // 16

// ---------------------------------------------------------------------------
// Kernel 0: init row/col min arrays in workspace to +inf (bit pattern).
// ---------------------------------------------------------------------------
__global__ void chamfer_init(unsigned int* __restrict__ ws, int n) {
    int i = blockIdx.x * blockDim.x + threadIdx.x;
    if (i < n) ws[i] = 0x7F800000u;  // +inf
}

// ---------------------------------------------------------------------------
// Kernel 1: 8 waves/block, one 16-row tile per wave.
// Phase 1: async DMA pred[b] (16 KB) into LDS (gfx1250 global->LDS async).
// Phase 2: cooperative pack pass -> WMMA-ready B columns in LDS:
//            pk[col] = [-2*y0, -2*y1, 1, |y|^2]   (float4, 32 KB)
// Phase 3: each wave sweeps 128 column tiles; per tile:
//            ds_load_b64 (conflict-free) -> V_WMMA_F32_16X16X4_F32 ->
//            8-way min tree -> uint atomicMin on column slot.
// A row = [x0, x1, |x|^2, 1], so one WMMA = 16x16 tile of squared distances.
// ---------------------------------------------------------------------------
__global__ __launch_bounds__(256) void chamfer_tiles(
    const float* __restrict__ pds,        // [B, N, 2]
    const float* __restrict__ pred,       // [B, M, 2]
    float* __restrict__ rowmin,           // [B, N]
    unsigned int* __restrict__ colmin)    // [B, M] (f32 bits)
{
    __shared__ float4 pk[M_];             // 32 KB: packed B columns
    __shared__ float2 raw[M_];            // 16 KB: async staging of pred[b]

    const int b    = blockIdx.x / BLOCKS_PER_BATCH;
    const int grp  = blockIdx.x % BLOCKS_PER_BATCH;
    const int w    = threadIdx.x >> 5;    // wave in block: 0..7
    const int lane = threadIdx.x & 31;    // 0..31 (wave32)
    const int l16  = lane & 15;
    const bool hi  = lane >= 16;          // upper half-wave holds K=2,3
    const int nt   = grp * WAVES_PER_BLOCK + w;

    // ---- Phase 1: async stage pred[b] -> LDS ----
    {
        const uint64_t gbase   = (uint64_t)(const void*)(pred + (size_t)b * M_ * 2);
        const unsigned ldsbase = (unsigned)(uintptr_t)(void*)&raw[0];
        for (int e = threadIdx.x; e < M_; e += 256) {
            unsigned goff = (unsigned)e * 8u;           // byte offset (float2)
            unsigned loff = ldsbase + (unsigned)e * 8u; // LDS byte address
            asm volatile("global_load_async_to_lds_b64 %0, %1, %2"
                         :
                         : "v"(loff), "v"(goff), "s"(gbase)
                         : "memory");
        }
        asm volatile("s_wait_asynccnt 0x0" ::: "memory");
    }
    __syncthreads();

    // ---- Phase 2: pack B columns once per block ----
    for (int e = threadIdx.x; e < M_; e += 256) {
        const float2 q = raw[e];
        pk[e] = make_float4(-2.0f * q.x, -2.0f * q.y, 1.0f, q.x * q.x + q.y * q.y);
    }
    __syncthreads();

    // ---- A tile: rows n0..n0+15 of pds[b], loop-invariant over M ----
    const int n0 = nt * 16;
    const float2 p = ((const float2*)(pds + (size_t)b * N_ * 2))[n0 + l16];
    v2f a;
    a.x = hi ? (p.x * p.x + p.y * p.y) : p.x;   // K=2 (|x|^2) : K=0 (x0)
    a.y = hi ? 1.0f                    : p.y;   // K=3 (1)     : K=1 (x1)

    float rmin[8];
#pragma unroll
    for (int r = 0; r < 8; ++r) rmin[r] = __builtin_inff();

    unsigned int* cmbase = colmin + (size_t)b * M_;

    // Per-lane view of packed columns: two float2 halves per column.
    // lo lanes read floats {0,1} (K=0,1), hi lanes floats {2,3} (K=2,3):
    // byte addr = col*16 + hi*8 -> all 64 LDS banks distinct, conflict-free,
    // and the layout matches the WMMA B operand directly.
    const float2* pk2 = (const float2*)pk;
    const int bofs = 2 * l16 + (hi ? 1 : 0);

    for (int mt = 0; mt < MT; mt += 4) {
        // Hoisted loads: 4 tiles in flight so dscnt waits pipeline.
        const float2 q0 = pk2[32 * (mt + 0) + bofs];
        const float2 q1 = pk2[32 * (mt + 1) + bofs];
        const float2 q2 = pk2[32 * (mt + 2) + bofs];
        const float2 q3 = pk2[32 * (mt + 3) + bofs];

#pragma unroll
        for (int j = 0; j < 4; ++j) {
            const float2 qj = (j == 0) ? q0 : (j == 1) ? q1 : (j == 2) ? q2 : q3;
            v2f bb; bb.x = qj.x; bb.y = qj.y;

            v8f c = {0.f, 0.f, 0.f, 0.f, 0.f, 0.f, 0.f, 0.f};
            v8f d = __builtin_amdgcn_wmma_f32_16x16x4_f32(
                /*neg_a=*/false, a, /*neg_b=*/false, bb,
                /*c_mod=*/(short)0, c, /*reuse_a=*/false, /*reuse_b=*/false);

            float cmin = d[0];
            rmin[0] = fminf(rmin[0], d[0]);
#pragma unroll
            for (int r = 1; r < 8; ++r) {
                rmin[r] = fminf(rmin[r], d[r]);
                cmin    = fminf(cmin, d[r]);
            }
            // lo lanes: min of rows 0-7; hi lanes: min of rows 8-15; both
            // merge into the same column slot. uint min on non-negative f32
            // bits == float min; order-invariant => deterministic.
            atomicMin(&cmbase[(mt + j) * 16 + l16],
                      __float_as_uint(fmaxf(cmin, 0.0f)));
        }
    }

    // Reduce row mins across the 16 lanes of each half-wave group.
#pragma unroll
    for (int s = 8; s >= 1; s >>= 1) {
#pragma unroll
        for (int r = 0; r < 8; ++r)
            rmin[r] = fminf(rmin[r], __shfl_xor(rmin[r], s, 32));
    }
    // Lane 0: rows n0+0..7 ; lane 16: rows n0+8..15. Each wave owns its rows.
    if (lane == 0) {
#pragma unroll
        for (int r = 0; r < 8; ++r)
            rowmin[(size_t)b * N_ + n0 + r] = fmaxf(rmin[r], 0.0f);
    }
    if (lane == 16) {
#pragma unroll
        for (int r = 0; r < 8; ++r)
            rowmin[(size_t)b * N_ + n0 + 8 + r] = fmaxf(rmin[r], 0.0f);
    }
}

// ---------------------------------------------------------------------------
// Kernel 2: deterministic single-block reduction.
// result = 0.5 * mean_b( mean_n sqrt(rowmin) + mean_m sqrt(colmin) )
//        = (sum_row + sum_col) / (2*B*N)   since N == M.
// ---------------------------------------------------------------------------
__global__ __launch_bounds__(256) void chamfer_reduce(
    const float* __restrict__ rowmin,
    const float* __restrict__ colmin,
    float* __restrict__ out)
{
    __shared__ float sdata[256];
    float s = 0.0f;
    for (int i = threadIdx.x; i < B_ * N_; i += 256) s += sqrtf(rowmin[i]);
    for (int i = threadIdx.x; i < B_ * M_; i += 256) s += sqrtf(colmin[i]);
    sdata[threadIdx.x] = s;
    __syncthreads();
#pragma unroll
    for (int k = 128; k >= 1; k >>= 1) {
        if (threadIdx.x < k) sdata[threadIdx.x] += sdata[threadIdx.x + k];
        __syncthreads();
    }
    if (threadIdx.x == 0)
        out[0] = sdata[0] / (float)(2 * B_ * N_);
}

// ---------------------------------------------------------------------------
extern "C" void kernel_launch(void* const* d_in, const int* in_sizes, int n_in,
                              void* d_out, int out_size, void* d_ws, size_t ws_size,
                              hipStream_t stream) {
    (void)in_sizes; (void)n_in; (void)out_size; (void)ws_size;
    const float* pds  = (const float*)d_in[0];   // [8, 2048, 2] f32
    const float* pred = (const float*)d_in[1];   // [8, 2048, 2] f32

    float*        rowmin = (float*)d_ws;                       // B*N floats
    unsigned int* colmin = (unsigned int*)(rowmin + B_ * N_);  // B*M floats (bits)

    const int n_ws = B_ * (N_ + M_);
    chamfer_init<<<(n_ws + 255) / 256, 256, 0, stream>>>((unsigned int*)d_ws, n_ws);
    chamfer_tiles<<<B_ * BLOCKS_PER_BATCH, 256, 0, stream>>>(pds, pred, rowmin, colmin);
    chamfer_reduce<<<1, 256, 0, stream>>>(rowmin, (const float*)colmin, (float*)d_out);
}